// MinkowskiUnion_23725399343383
// MI455X (gfx1250) — compile-verified
//
#include <hip/hip_runtime.h>

// MinkowskiEngine union forward: out = zeros(n_out, 64); out[idx1]+=feats1; out[idx2]+=feats2
//
// Memory-bound scatter-add (AI ~0.09 FLOP/B -> ~30us at 23.3 TB/s). MI455X plan:
//  - output (192 MB == global L2 size) stays L2-resident; atomics resolve at L2
//  - inputs streamed with non-temporal (TH=NT) loads so one-shot traffic
//    doesn't evict the output working set
//  - fp32 no-return global atomics (STOREcnt-tracked, no pre-op return stall)
//  - wave32 layout: one wave per 64-channel row; row index forced scalar via
//    readfirstlane so idx comes in on the SMEM/constant-cache path (s_load)

typedef float v2f __attribute__((ext_vector_type(2)));
typedef float v4f __attribute__((ext_vector_type(4)));

#define CHANNELS 64

__global__ void union_scatter_add_kernel(const float* __restrict__ feats1,
                                         const float* __restrict__ feats2,
                                         const int*   __restrict__ idx1,
                                         const int*   __restrict__ idx2,
                                         float*       __restrict__ out,
                                         int n1, int n2) {
    int gid  = blockIdx.x * blockDim.x + threadIdx.x;
    int lane = threadIdx.x & 31;                 // 2 channels per lane
    // wave-uniform row id, forced into an SGPR -> scalar branch + s_load for idx
    int row  = __builtin_amdgcn_readfirstlane(gid >> 5);
    if (row >= n1 + n2) return;

    const float* src;
    int orow;
    if (row < n1) {
        src  = feats1 + (size_t)row * CHANNELS;
        orow = idx1[row];                        // uniform address -> s_load_b32
    } else {
        int r = row - n1;
        src  = feats2 + (size_t)r * CHANNELS;
        orow = idx2[r];
    }

    // one-use streaming data: NT hint preserves L2 for the output array
    v2f v = __builtin_nontemporal_load((const v2f*)src + lane);

    float* dst = out + (size_t)orow * CHANNELS + lane * 2;
    // result discarded -> global_atomic_add_f32 without return (STOREcnt)
    atomicAdd(dst,     v.x);
    atomicAdd(dst + 1, v.y);
}

// Each thread writes 4 float4s (64B), block-chunked so every b128 store is
// lane-contiguous across the wave. out_size is a multiple of 4 (C=64), so the
// per-float4 guard covers the range exactly; no scalar tail needed.
__global__ void union_zero_kernel(float* __restrict__ out, int n4) {
    int base  = blockIdx.x * (blockDim.x * 4) + threadIdx.x;
    v4f* out4 = (v4f*)out;
#pragma unroll
    for (int k = 0; k < 4; ++k) {
        int j = base + k * blockDim.x;
        if (j < n4) out4[j] = (v4f){0.f, 0.f, 0.f, 0.f};
    }
}

extern "C" void kernel_launch(void* const* d_in, const int* in_sizes, int n_in,
                              void* d_out, int out_size, void* d_ws, size_t ws_size,
                              hipStream_t stream) {
    const float* feats1 = (const float*)d_in[0];
    const float* feats2 = (const float*)d_in[1];
    const int*   idx1   = (const int*)d_in[2];
    const int*   idx2   = (const int*)d_in[3];
    // d_in[4] is n_out on device; out_size == n_out * 64 already

    const int n1 = in_sizes[2];   // rows in feats1
    const int n2 = in_sizes[3];   // rows in feats2
    float* out = (float*)d_out;

    // 1) zero the union output (harness poisons d_out; must re-zero every call)
    {
        int n4      = out_size >> 2;            // float4 count (out_size % 4 == 0)
        int perBlk  = 256 * 4;                  // float4s per block
        int blocks  = (n4 + perBlk - 1) / perBlk;
        if (blocks < 1) blocks = 1;
        union_zero_kernel<<<blocks, 256, 0, stream>>>(out, n4);
    }

    // 2) scatter-add both inputs in one launch (stream order guarantees the
    //    zero-fill completed first)
    {
        long long total = (long long)(n1 + n2) * 32;   // 32 lanes per row
        int blocks = (int)((total + 255) / 256);
        union_scatter_add_kernel<<<blocks, 256, 0, stream>>>(
            feats1, feats2, idx1, idx2, out, n1, n2);
    }
}